// ComplexGAT_79731772883559
// MI455X (gfx1250) — compile-verified
//
#include <hip/hip_runtime.h>
#include <math.h>

#define NNODES 50000
#define NEDGES 800000
#define ETOT   (NEDGES + NNODES)   // 850000, divisible by 8
#define HC     128
#define NHEAD  4
#define GB     8
#define NCLS   10

typedef __attribute__((ext_vector_type(2))) float v2f;
typedef __attribute__((ext_vector_type(8))) float v8f;

__device__ __forceinline__ float leaky02(float t) { return t > 0.f ? t : 0.2f * t; }

// order-preserving float<->uint encoding for atomicMax-based segment max
__device__ __forceinline__ unsigned fenc(float f) {
  unsigned u = __float_as_uint(f);
  return (u & 0x80000000u) ? ~u : (u | 0x80000000u);
}
__device__ __forceinline__ float fdec(unsigned u) {
  return (u & 0x80000000u) ? __uint_as_float(u & 0x7FFFFFFFu) : __uint_as_float(~u);
}

// ---------------------------------------------------------------------------
// GEMM: H = A[N,128] @ W[128,128] using V_WMMA_F32_16X16X4_F32 (full f32).
// One wave per 16x16 output tile; 8 waves/block -> 16x128 stripe per block.
// A tile staged in LDS (8 KB); W stays L2-resident (64 KB).
// ---------------------------------------------------------------------------
__global__ __launch_bounds__(256) void gat_gemm_wmma(const float* __restrict__ A,
                                                     const float* __restrict__ W,
                                                     float* __restrict__ H) {
  __shared__ float As[16 * HC];
  const int row0 = blockIdx.x << 4;
  const int t = threadIdx.x;
  {
    const float4* src = (const float4*)(A + (size_t)row0 * HC);
    float4* dst = (float4*)As;
    for (int i = t; i < (16 * HC) / 4; i += 256) dst[i] = src[i];
  }
  __syncthreads();

  const int lane = t & 31;
  const int col0 = (t >> 5) << 4;        // wave id * 16
  const int m    = lane & 15;            // row (A) / col (B,C) within tile
  const int kh   = (lane >> 4) << 1;     // 0 for lanes 0-15, 2 for lanes 16-31

  v8f c = {};
  for (int k0 = 0; k0 < HC; k0 += 4) {
    v2f a, b;
    // A 16x4 f32 layout: lanes 0-15 hold K=k0,k0+1; lanes 16-31 hold K=k0+2,k0+3
    a.x = As[m * HC + k0 + kh];
    a.y = As[m * HC + k0 + kh + 1];
    // B 4x16 f32 layout (symmetric): lane n holds column n, same K split
    b.x = W[(size_t)(k0 + kh) * HC + col0 + m];
    b.y = W[(size_t)(k0 + kh + 1) * HC + col0 + m];
    c = __builtin_amdgcn_wmma_f32_16x16x4_f32(false, a, false, b, (short)0, c,
                                              false, false);
  }
  // C 16x16 f32 layout: VGPR j -> M=j (lanes 0-15) / M=j+8 (lanes 16-31), N=lane&15
  const int rbase = row0 + ((lane >> 4) << 3);
  for (int j = 0; j < 8; ++j)
    H[(size_t)(rbase + j) * HC + col0 + m] = c[j];
}

// ---------------------------------------------------------------------------
// Per-node attention logits al_src/al_dst [N,H]; also init m_enc, z, agg=0.
// One wave per node; lane l owns channels 4l..4l+3 (head = l>>3).
// ---------------------------------------------------------------------------
__global__ __launch_bounds__(256) void gat_node_prep(
    const float* __restrict__ H, const float* __restrict__ att_src,
    const float* __restrict__ att_dst, float* __restrict__ al_src,
    float* __restrict__ al_dst, unsigned* __restrict__ m_enc,
    float* __restrict__ z, float* __restrict__ agg) {
  const int node = blockIdx.x * 8 + (threadIdx.x >> 5);
  const int lane = threadIdx.x & 31;
  if (node >= NNODES) return;
  float4 v  = ((const float4*)(H + (size_t)node * HC))[lane];
  float4 as = ((const float4*)att_src)[lane];
  float4 ad = ((const float4*)att_dst)[lane];
  float s = v.x * as.x + v.y * as.y + v.z * as.z + v.w * as.w;
  float d = v.x * ad.x + v.y * ad.y + v.z * ad.z + v.w * ad.w;
  // reduce across the 8 lanes of each head group
  s += __shfl_xor(s, 1, 32);  d += __shfl_xor(d, 1, 32);
  s += __shfl_xor(s, 2, 32);  d += __shfl_xor(d, 2, 32);
  s += __shfl_xor(s, 4, 32);  d += __shfl_xor(d, 4, 32);
  if ((lane & 7) == 0) {
    const int hh = lane >> 3;
    al_src[node * NHEAD + hh] = s;
    al_dst[node * NHEAD + hh] = d;
    m_enc[node * NHEAD + hh]  = 0u;   // below every encoded float
    z[node * NHEAD + hh]      = 0.f;
  }
  ((float4*)(agg + (size_t)node * HC))[lane] = float4{0.f, 0.f, 0.f, 0.f};
}

__device__ __forceinline__ void edge_endpoints(const int* __restrict__ ei, int e,
                                               int& s, int& d) {
  if (e < NEDGES) { s = ei[e]; d = ei[NEDGES + e]; }
  else            { s = e - NEDGES; d = s; }        // self loops
}

// pass 1: segment max of LeakyReLU logits via encoded atomicMax
__global__ __launch_bounds__(256) void gat_edge_max(
    const int* __restrict__ ei, const float* __restrict__ al_src,
    const float* __restrict__ al_dst, unsigned* __restrict__ m_enc) {
  const int e = blockIdx.x * 256 + threadIdx.x;
  if (e >= ETOT) return;
  int s, d; edge_endpoints(ei, e, s, d);
  float4 ls = *(const float4*)(al_src + (size_t)s * NHEAD);
  float4 ld = *(const float4*)(al_dst + (size_t)d * NHEAD);
  unsigned* md = m_enc + (size_t)d * NHEAD;
  atomicMax(md + 0, fenc(leaky02(ls.x + ld.x)));
  atomicMax(md + 1, fenc(leaky02(ls.y + ld.y)));
  atomicMax(md + 2, fenc(leaky02(ls.z + ld.z)));
  atomicMax(md + 3, fenc(leaky02(ls.w + ld.w)));
}

// pass 2: segment sum of exp(e - m)
__global__ __launch_bounds__(256) void gat_edge_sum(
    const int* __restrict__ ei, const float* __restrict__ al_src,
    const float* __restrict__ al_dst, const unsigned* __restrict__ m_enc,
    float* __restrict__ z) {
  const int e = blockIdx.x * 256 + threadIdx.x;
  if (e >= ETOT) return;
  int s, d; edge_endpoints(ei, e, s, d);
  float4 ls = *(const float4*)(al_src + (size_t)s * NHEAD);
  float4 ld = *(const float4*)(al_dst + (size_t)d * NHEAD);
  const unsigned* md = m_enc + (size_t)d * NHEAD;
  float* zd = z + (size_t)d * NHEAD;
  atomicAdd(zd + 0, expf(leaky02(ls.x + ld.x) - fdec(md[0])));
  atomicAdd(zd + 1, expf(leaky02(ls.y + ld.y) - fdec(md[1])));
  atomicAdd(zd + 2, expf(leaky02(ls.z + ld.z) - fdec(md[2])));
  atomicAdd(zd + 3, expf(leaky02(ls.w + ld.w) - fdec(md[3])));
}

// pass 3 (dominant, bandwidth-bound): agg[dst] += h[src] * alpha
// one wave per edge; lane l -> channels 4l..4l+3 (head l>>3); 512B coalesced/edge
__global__ __launch_bounds__(256) void gat_edge_aggr(
    const int* __restrict__ ei, const float* __restrict__ al_src,
    const float* __restrict__ al_dst, const unsigned* __restrict__ m_enc,
    const float* __restrict__ z, const float* __restrict__ H,
    float* __restrict__ agg) {
  const int e = blockIdx.x * 8 + (threadIdx.x >> 5);
  if (e >= ETOT) return;
  const int lane = threadIdx.x & 31;
  int s, d; edge_endpoints(ei, e, s, d);
  const int hh = lane >> 3;
  float t = leaky02(al_src[(size_t)s * NHEAD + hh] + al_dst[(size_t)d * NHEAD + hh]);
  float alpha = expf(t - fdec(m_enc[(size_t)d * NHEAD + hh])) /
                z[(size_t)d * NHEAD + hh];
  float4 v = ((const float4*)(H + (size_t)s * HC))[lane];
  float* o = agg + (size_t)d * HC + lane * 4;
  atomicAdd(o + 0, v.x * alpha);
  atomicAdd(o + 1, v.y * alpha);
  atomicAdd(o + 2, v.z * alpha);
  atomicAdd(o + 3, v.w * alpha);
}

__global__ __launch_bounds__(256) void gat_bias_elu(float* __restrict__ agg,
                                                    const float* __restrict__ bias) {
  const int i = blockIdx.x * 256 + threadIdx.x;   // over N*HC exactly
  float v = agg[i] + bias[i & (HC - 1)];
  agg[i] = v > 0.f ? v : expm1f(v);
}

__global__ void gat_pool_init(float* __restrict__ sums, float* __restrict__ cnt) {
  const int t = threadIdx.x;
  for (int i = t; i < GB * HC; i += 256) sums[i] = 0.f;
  if (t < GB) cnt[t] = 0.f;
}

__global__ __launch_bounds__(256) void gat_pool(const float* __restrict__ feat,
                                                const int* __restrict__ batch,
                                                float* __restrict__ sums,
                                                float* __restrict__ cnt) {
  const int node = blockIdx.x * 8 + (threadIdx.x >> 5);
  const int lane = threadIdx.x & 31;
  if (node >= NNODES) return;
  const int g = batch[node];
  float4 v = ((const float4*)(feat + (size_t)node * HC))[lane];
  float* o = sums + (size_t)g * HC + lane * 4;
  atomicAdd(o + 0, v.x); atomicAdd(o + 1, v.y);
  atomicAdd(o + 2, v.z); atomicAdd(o + 3, v.w);
  if (lane == 0) atomicAdd(cnt + g, 1.f);
}

__global__ __launch_bounds__(128) void gat_head(const float* __restrict__ sums,
                                                const float* __restrict__ cnt,
                                                const float* __restrict__ linW,
                                                const float* __restrict__ linb,
                                                float* __restrict__ out) {
  __shared__ float pooled[GB * HC];
  __shared__ float logits[GB * NCLS];
  const int t = threadIdx.x;
  for (int i = t; i < GB * HC; i += 128) {
    const int g = i >> 7;
    float c = cnt[g]; if (c < 1.f) c = 1.f;
    pooled[i] = sums[i] / c;
  }
  __syncthreads();
  if (t < GB * NCLS) {
    const int g = t / NCLS, k = t % NCLS;
    float acc = linb[k];
    for (int j = 0; j < HC; ++j) acc += pooled[g * HC + j] * linW[j * NCLS + k];
    logits[t] = acc;
  }
  __syncthreads();
  if (t < GB * NCLS) {
    const int g = t / NCLS;
    float mx = logits[g * NCLS];
    for (int j = 1; j < NCLS; ++j) mx = fmaxf(mx, logits[g * NCLS + j]);
    float ss = 0.f;
    for (int j = 0; j < NCLS; ++j) ss += expf(logits[g * NCLS + j] - mx);
    out[t] = logits[t] - mx - logf(ss);
  }
}

extern "C" void kernel_launch(void* const* d_in, const int* in_sizes, int n_in,
                              void* d_out, int out_size, void* d_ws, size_t ws_size,
                              hipStream_t stream) {
  (void)in_sizes; (void)n_in; (void)out_size; (void)ws_size;
  const float* x     = (const float*)d_in[0];
  const int*   ei    = (const int*)d_in[1];
  const int*   batch = (const int*)d_in[2];
  const float *Wl[4], *asl[4], *adl[4], *bl[4];
  for (int L = 0; L < 4; ++L) {
    Wl[L]  = (const float*)d_in[3 + 4 * L];
    asl[L] = (const float*)d_in[4 + 4 * L];
    adl[L] = (const float*)d_in[5 + 4 * L];
    bl[L]  = (const float*)d_in[6 + 4 * L];
  }
  const float* linW = (const float*)d_in[19];
  const float* linb = (const float*)d_in[20];

  char* ws = (char*)d_ws;
  size_t off = 0;
  auto take = [&](size_t bytes) -> char* {
    char* p = ws + off;
    off = (off + bytes + 255) & ~(size_t)255;
    return p;
  };
  float*    hbuf   = (float*)take((size_t)NNODES * HC * 4);
  float*    aggA   = (float*)take((size_t)NNODES * HC * 4);
  float*    aggB   = (float*)take((size_t)NNODES * HC * 4);
  float*    al_src = (float*)take((size_t)NNODES * NHEAD * 4);
  float*    al_dst = (float*)take((size_t)NNODES * NHEAD * 4);
  unsigned* m_enc  = (unsigned*)take((size_t)NNODES * NHEAD * 4);
  float*    z      = (float*)take((size_t)NNODES * NHEAD * 4);
  float*    sums   = (float*)take(GB * HC * 4);
  float*    cnt    = (float*)take(GB * 4);

  const float* in_feat = x;
  float* aggs[2] = {aggA, aggB};
  const int egrid = (ETOT + 255) / 256;
  for (int L = 0; L < 4; ++L) {
    float* agg = aggs[L & 1];
    gat_gemm_wmma<<<NNODES / 16, 256, 0, stream>>>(in_feat, Wl[L], hbuf);
    gat_node_prep<<<NNODES / 8, 256, 0, stream>>>(hbuf, asl[L], adl[L], al_src,
                                                  al_dst, m_enc, z, agg);
    gat_edge_max<<<egrid, 256, 0, stream>>>(ei, al_src, al_dst, m_enc);
    gat_edge_sum<<<egrid, 256, 0, stream>>>(ei, al_src, al_dst, m_enc, z);
    gat_edge_aggr<<<ETOT / 8, 256, 0, stream>>>(ei, al_src, al_dst, m_enc, z,
                                                hbuf, agg);
    gat_bias_elu<<<(NNODES * HC) / 256, 256, 0, stream>>>(agg, bl[L]);
    in_feat = agg;
  }
  gat_pool_init<<<1, 256, 0, stream>>>(sums, cnt);
  gat_pool<<<NNODES / 8, 256, 0, stream>>>(in_feat, batch, sums, cnt);
  gat_head<<<1, 128, 0, stream>>>(sums, cnt, linW, linb, (float*)d_out);
}